// ConvGroupShiftTaichi_62388694941882
// MI455X (gfx1250) — compile-verified
//
#include <hip/hip_runtime.h>

typedef float v2f __attribute__((ext_vector_type(2)));
typedef float v8f __attribute__((ext_vector_type(8)));
typedef unsigned int u32x4 __attribute__((ext_vector_type(4)));
typedef int i32x4 __attribute__((ext_vector_type(4)));
typedef int i32x8 __attribute__((ext_vector_type(8)));

#define B_    8
#define CIN   256
#define COUT  256
#define H_    64
#define W_    64
#define CBLK  64      // out-channels per workgroup
#define IB    8       // output rows per workgroup
#define CCH   8       // cin per K-chunk
#define KCH   40      // CCH * 5 (kw)
#define TL    11      // taps per t-range (5 ranges cover t in [0,55))
#define XROWS 18      // IB + TL - 1
#define XCOLS 68      // 64 + 4 halo

#if __has_builtin(__builtin_amdgcn_tensor_load_to_lds)
#define USE_TDM 1
#if __has_include(<hip/amd_detail/amd_gfx1250_TDM.h>)
#define TDM_ARITY 6
#pragma message("CDNA5 probe: tensor_load_to_lds builtin PRESENT, TDM header present -> arity 6")
#else
#define TDM_ARITY 5
#pragma message("CDNA5 probe: tensor_load_to_lds builtin PRESENT, no TDM header -> arity 5")
#endif
#else
#define USE_TDM 0
#pragma message("CDNA5 probe: tensor_load_to_lds builtin ABSENT -> cooperative staging fallback")
#endif

#if USE_TDM
// TDM 3-D tile: x = 5 kw (contiguous), y = 64 out-channels (stride 70400),
// z = 8 cin (stride 25). Linear LDS fill order (z,y,x) => sA[ccl][c][kw].
__device__ __forceinline__ void tdm_issue_panel(const float* wbase, unsigned ldsOff)
{
    unsigned long long ga = (unsigned long long)(size_t)wbase;
    u32x4 g0;
    g0[0] = 1u;                                                // count=1, user mode
    g0[1] = ldsOff;                                            // lds_addr (bytes)
    g0[2] = (unsigned)(ga & 0xffffffffu);                      // global_addr lo
    g0[3] = (unsigned)((ga >> 32) & 0x01ffffffu) | (2u << 30); // global_addr hi | type=2
    i32x8 g1;
    g1[0] = (int)(2u << 16);                 // wg_mask=0, data_size=4B
    g1[1] = (int)(25u << 16);                // tensor_dim0[15:0] = 25
    g1[2] = 0;                               // dim0 hi | tensor_dim1 lo16 (dim1=0x100000)
    g1[3] = (int)(0x10u | (5u << 16));       // tensor_dim1 hi16 | tile_dim0=5
    g1[4] = (int)(64u | (8u << 16));         // tile_dim1=64 (ch), tile_dim2=8 (cin)
    g1[5] = 70400;                           // tensor_dim0_stride lo32 (ch stride)
    g1[6] = (int)(25u << 16);                // stride0 hi16=0 | stride1 lo16 = 25
    g1[7] = 0;                               // stride1 [47:16]
    i32x4 g2; g2[0] = 256; g2[1] = 0; g2[2] = 0; g2[3] = 0;    // tensor_dim2 >= 8
    i32x4 g3; g3[0] = 0;   g3[1] = 0; g3[2] = 0; g3[3] = 0;
#if TDM_ARITY == 6
    i32x8 gx = {0, 0, 0, 0, 0, 0, 0, 0};
    __builtin_amdgcn_tensor_load_to_lds(g0, g1, g2, g3, gx, 0);
#else
    __builtin_amdgcn_tensor_load_to_lds(g0, g1, g2, g3, 0);
#endif
}
#endif

// out[b,c,i,j] = sum_{cin, t in [0,55), kw} x[b,cin,i-25+t,j-2+kw] * W[c*11+t/5, cin, t%5, kw]
__global__ __launch_bounds__(256)
void conv_shift_wmma(const float* __restrict__ x, const float* __restrict__ w,
                     float* __restrict__ out, float* __restrict__ ws)
{
    __shared__ float sX[CCH][XROWS][XCOLS];   // 39,168 B
    __shared__ float sA[2][CCH][CBLK][5];     // 20,480 B (double buffer, cin-major)

    const int tid  = threadIdx.x;
    const int lane = tid & 31;
    const int wv   = tid >> 5;             // wave id == output row within block
    const int lm   = lane & 15;            // m (A) / n (B) within 16x16 tile
    const int kbl  = (lane >> 4) << 1;     // per-lane K base: 0 or 2
    const int c0   = blockIdx.x * CBLK;
    const int i0   = blockIdx.y * IB;
    const int b    = blockIdx.z;

    v8f acc[4][4];                         // 4 M-tiles x 4 N-tiles per wave
    #pragma unroll
    for (int mt = 0; mt < 4; ++mt)
        #pragma unroll
        for (int nt = 0; nt < 4; ++nt)
            #pragma unroll
            for (int e = 0; e < 8; ++e)
                acc[mt][nt][e] = 0.0f;

    for (int r = 0; r < 5; ++r) {
        const int tlo = r * TL;
        int tA = tlo, tB = tlo + TL;              // valid t: i0+t in [18,88]
        if (tA < 18 - i0) tA = 18 - i0;
        if (tB > 89 - i0) tB = 89 - i0;
        if (tA >= tB) continue;                   // uniform per block
        const int nT = tB - tA;

        for (int cc0 = 0; cc0 < CIN; cc0 += CCH) {
            __syncthreads();   // previous readers of sX / sA[0] done
#if USE_TDM
            if (wv == 0) {     // prologue panel overlaps with x staging below
                const int idx0 = tA / 5, kh0 = tA % 5;
                tdm_issue_panel(w + (((size_t)c0 * 11 + idx0) * CIN + cc0) * 25 + kh0 * 5,
                                (unsigned)(size_t)&sA[0][0][0][0]);
            }
#endif
            // ---- stage x strip (manual: needs left/top zero halo)
            for (int u = tid; u < CCH * XROWS * XCOLS; u += 256) {
                int cc  = u / (XROWS * XCOLS);
                int rm  = u - cc * (XROWS * XCOLS);
                int rr  = rm / XCOLS;
                int col = rm - rr * XCOLS;
                int grow = i0 - 25 + tlo + rr;
                int gcol = col - 2;
                float v = 0.0f;
                if ((unsigned)grow < 64u && (unsigned)gcol < 64u)
                    v = x[(((b * CIN) + cc0 + cc) * H_ + grow) * W_ + gcol];
                sX[cc][rr][col] = v;
            }
            __syncthreads();

            int buf = 0;
            for (int v5 = 0; v5 < nT; ++v5) {
                const int t  = tA + v5;
                const int tt = t - tlo;
#if USE_TDM
                if (wv == 0) {
                    if (v5 + 1 < nT) {            // issue next panel, wait current
                        const int tn = t + 1;
                        const int idxn = tn / 5, khn = tn % 5;
                        tdm_issue_panel(w + (((size_t)c0 * 11 + idxn) * CIN + cc0) * 25 + khn * 5,
                                        (unsigned)(size_t)&sA[buf ^ 1][0][0][0]);
                        __builtin_amdgcn_s_wait_tensorcnt((short)1);
                    } else {
                        __builtin_amdgcn_s_wait_tensorcnt((short)0);
                    }
                }
#else
                {   // cooperative weight panel staging: sA[buf][ccl][c][kw]
                    const int idx = t / 5, kh = t % 5;
                    for (int u = tid; u < CCH * CBLK * 5; u += 256) {
                        int ccl = u / (CBLK * 5);
                        int rm2 = u - ccl * (CBLK * 5);
                        int c   = rm2 / 5, kw2 = rm2 - c * 5;
                        sA[buf][ccl][c][kw2] =
                            w[(((size_t)(c0 + c) * 11 + idx) * CIN + cc0 + ccl) * 25 + kh * 5 + kw2];
                    }
                }
#endif
                __syncthreads();

                const int row = wv + tt;          // this wave's x row in sX
                // ---- K' = 40 : ten K=4 WMMA steps, 16 tiles per wave
                #pragma unroll
                for (int ks = 0; ks < 10; ++ks) {
                    const int ka  = ks * 4 + kbl;
                    const int kb2 = ka + 1;
                    const int ca = ka / 5,  kwa = ka - ca * 5;
                    const int cb = kb2 / 5, kwb = kb2 - cb * 5;
                    v2f a[4];
                    #pragma unroll
                    for (int mt = 0; mt < 4; ++mt) {
                        a[mt].x = sA[buf][ca][mt * 16 + lm][kwa];
                        a[mt].y = sA[buf][cb][mt * 16 + lm][kwb];
                    }
                    #pragma unroll
                    for (int nt = 0; nt < 4; ++nt) {
                        const int jb = nt * 16;
                        v2f bf;
                        bf.x = sX[ca][row][jb + lm + kwa];
                        bf.y = sX[cb][row][jb + lm + kwb];
                        #pragma unroll
                        for (int mt = 0; mt < 4; ++mt)
                            acc[mt][nt] = __builtin_amdgcn_wmma_f32_16x16x4_f32(
                                false, a[mt], false, bf, (short)0, acc[mt][nt], false, false);
                    }
                }
                __syncthreads();   // all reads of sA[buf]/sX done before next overwrite
                buf ^= 1;
            }
        }
    }

    // ---- epilogue: store tile + deterministic per-channel partials (two passes)
    float* sP = &sX[0][0][0];      // alias, sX dead: 64 ch * 128 slots = 8192 floats
    const int slot = wv * 16 + lm;
    const int h    = lane >> 4;
    const int g    = blockIdx.y * 8 + b;
    __syncthreads();
    // pass 1: outputs + sums
    #pragma unroll
    for (int mt = 0; mt < 4; ++mt) {
        #pragma unroll
        for (int e = 0; e < 8; ++e) {
            const int cl = mt * 16 + e + (h << 3);   // C/D layout: M = e (+8 hi half)
            const int c  = c0 + cl;
            float s = 0.0f;
            #pragma unroll
            for (int nt = 0; nt < 4; ++nt) {
                float v = acc[mt][nt][e];
                out[(((b * COUT) + c) * H_ + i0 + wv) * W_ + nt * 16 + lm] = v;
                s += v;
            }
            sP[cl * 128 + slot] = s;
        }
    }
    __syncthreads();
    if (tid < CBLK) {
        float s = 0.0f;
        for (int g2 = 0; g2 < 128; ++g2) s += sP[tid * 128 + g2];
        ws[(c0 + tid) * 64 + g] = s;
    }
    __syncthreads();
    // pass 2: sums of squares
    #pragma unroll
    for (int mt = 0; mt < 4; ++mt) {
        #pragma unroll
        for (int e = 0; e < 8; ++e) {
            const int cl = mt * 16 + e + (h << 3);
            float q = 0.0f;
            #pragma unroll
            for (int nt = 0; nt < 4; ++nt) {
                float v = acc[mt][nt][e];
                q += v * v;
            }
            sP[cl * 128 + slot] = q;
        }
    }
    __syncthreads();
    if (tid < CBLK) {
        float q = 0.0f;
        for (int g2 = 0; g2 < 128; ++g2) q += sP[tid * 128 + g2];
        ws[16384 + (c0 + tid) * 64 + g] = q;
    }
}

__global__ void bn_finalize(const float* __restrict__ gamma, const float* __restrict__ beta,
                            float* __restrict__ ws)
{
    int c = threadIdx.x;                 // 256 threads, 1 block
    float s = 0.0f, q = 0.0f;
    for (int g = 0; g < 64; ++g) { s += ws[c * 64 + g]; q += ws[16384 + c * 64 + g]; }
    const float inv = 1.0f / 32768.0f;   // B*H*W
    float mean = s * inv;
    float var  = q * inv - mean * mean;  // population var (ddof=0)
    float sc   = gamma[c] * rsqrtf(var + 1e-5f);
    ws[32768 + c] = sc;
    ws[33024 + c] = beta[c] - mean * sc;
}

__global__ void bn_apply(float4* __restrict__ out4, const float* __restrict__ ws)
{
    unsigned i = blockIdx.x * 256u + threadIdx.x;   // 2,097,152 float4s
    int c = (i >> 10) & 255;                        // 1024 float4 per channel
    float sc = ws[32768 + c], sh = ws[33024 + c];
    float4 v = out4[i];
    v.x = v.x * sc + sh; v.y = v.y * sc + sh;
    v.z = v.z * sc + sh; v.w = v.w * sc + sh;
    out4[i] = v;
}

extern "C" void kernel_launch(void* const* d_in, const int* in_sizes, int n_in,
                              void* d_out, int out_size, void* d_ws, size_t ws_size,
                              hipStream_t stream)
{
    const float* x     = (const float*)d_in[0];
    const float* w     = (const float*)d_in[1];
    const float* gamma = (const float*)d_in[2];
    const float* beta  = (const float*)d_in[3];
    float* out = (float*)d_out;
    float* ws  = (float*)d_ws;

    dim3 grid(COUT / CBLK, H_ / IB, B_);   // 4 x 8 x 8 = 256 workgroups
    conv_shift_wmma<<<grid, 256, 0, stream>>>(x, w, out, ws);
    bn_finalize<<<1, 256, 0, stream>>>(gamma, beta, ws);
    bn_apply<<<(B_ * COUT * H_ * W_) / (256 * 4), 256, 0, stream>>>((float4*)out, ws);
}